// _TangentProject_15324443312268
// MI455X (gfx1250) — compile-verified
//
#include <hip/hip_runtime.h>
#include <hip/hip_bf16.h>

// ---------------------------------------------------------------------------
// TangentProject for MI455X (gfx1250): eigen-free, GEMM-only formulation.
//   sri      = W^{-1/2}        via coupled Newton-Schulz  (3 GEMMs / iter)
//   cong     = sri @ A @ sri   (2 GEMMs)
//   log(M)   = log(s)I + 2^k * log( (M/s)^{1/2^k} )  -- k repeated NS sqrts
//              + Horner/Taylor log(I+X)               (1 GEMM per term)
// All GEMMs run on V_WMMA_F32_16X16X4_F32 (full-precision matrix pipe).
// LDS layouts are chosen so every WMMA operand fragment is one aligned
// ds_load_b64 (no VALU marshalling between matrix ops).
// ---------------------------------------------------------------------------

typedef float v2f __attribute__((ext_vector_type(2)));
typedef float v8f __attribute__((ext_vector_type(8)));

#define MATN 256
#define MATE 65536L          // elements per 256x256 matrix

#define A_ROW 34             // even stride: 8B-aligned k-pairs, odd/17 bank walk
#define B_ROW 288            // 288 % 64 == 32 -> half-wave kp/kp+1 rows on
                             // disjoint bank halves (0..31 vs 32..63)

// operand batch-offset modes for the batched GEMM
//  0 = chunk-local linear (lb * MATE)
//  1 = weight index  c = (globalPair & 7)
//  2 = input  index  b = (globalPair >> 3)
__device__ __forceinline__ long opOff(int mode, int gp, int lb) {
  switch (mode) {
    case 1:  return (long)(gp & 7)  * MATE;
    case 2:  return (long)(gp >> 3) * MATE;
    default: return (long)lb        * MATE;
  }
}

// ---------------------------------------------------------------------------
// Batched 256x256x256 f32 GEMM on WMMA:  C = alpha*(A@B) + (beta + bp[b]) * I
// Block = 256 threads (8 wave32), 64(M) x 128(N) tile, K staged in LDS by 32.
// Each wave owns a 32x32 slab -> 4 v8f accumulators, 4 WMMA per 4 b64 loads.
// ---------------------------------------------------------------------------
__global__ __launch_bounds__(256) void gemm256(
    const float* __restrict__ Ab, const float* __restrict__ Bb,
    float* __restrict__ Cb, int amode, int bmode, int p0,
    float alpha, float beta, const float* __restrict__ betaPtr)
{
  // As[m][k]       : 64 x 32 tile, row stride A_ROW (k-fastest)
  // Bs[kp][n][2]   : 32 x 128 tile, k-pairs interleaved so (B[2kp][n],
  //                  B[2kp+1][n]) is one contiguous 8-byte fragment
  __shared__ __align__(16) float As[64 * A_ROW];
  __shared__ __align__(16) float Bs[16 * B_ROW];

  const int lb = blockIdx.y;
  const int gp = p0 + lb;
  const float* __restrict__ A = Ab + opOff(amode, gp, lb);
  const float* __restrict__ B = Bb + opOff(bmode, gp, lb);
  float* __restrict__ C = Cb + (long)lb * MATE;

  const int tid   = threadIdx.x;
  const int lane  = tid & 31;
  const int wid   = tid >> 5;                 // 0..7
  const int tileM = (blockIdx.x & 3) * 64;    // 4 M-blocks
  const int tileN = (blockIdx.x >> 2) * 128;  // 2 N-blocks
  const int waveM = (wid >> 2) * 32;          // 0,32
  const int waveN = (wid & 3) * 32;           // 0,32,64,96
  const int lm    = lane & 15;
  const int hw    = lane >> 4;                // half-wave: 0 or 1
  const int koff  = hw * 2;                   // lanes 16..31 hold K+2,K+3

  v8f acc00 = {}, acc01 = {}, acc10 = {}, acc11 = {};

  const int kL  = tid & 31,  mB  = tid >> 5;  // A staging coords
  const int nL  = tid & 127, kb2 = tid >> 7;  // B staging coords (kb2: 0/1)

  for (int kb = 0; kb < MATN; kb += 32) {
#pragma unroll
    for (int i = 0; i < 8; ++i)               // 64x32 A tile, coalesced on k
      As[(mB + 8 * i) * A_ROW + kL] =
          A[(long)(tileM + mB + 8 * i) * MATN + kb + kL];
#pragma unroll
    for (int i = 0; i < 16; ++i) {            // 32x128 B tile, coalesced on n
      const int k = kb2 + 2 * i;              // k>>1 == i, k&1 == kb2
      Bs[i * B_ROW + nL * 2 + kb2] =
          B[(long)(kb + k) * MATN + tileN + nL];
    }
    __syncthreads();

#pragma unroll
    for (int kk = 0; kk < 32; kk += 4) {
      const int kp = (kk + koff) >> 1;
      // A frag 16x4: lanes 0-15 -> (M=lm, K=kk,kk+1), lanes 16-31 -> K+2,K+3
      const v2f a0 = *(const v2f*)&As[(waveM + lm) * A_ROW + kk + koff];
      const v2f a1 = *(const v2f*)&As[(waveM + 16 + lm) * A_ROW + kk + koff];
      // B frag 4x16: VGPR0 rows K / K+2, VGPR1 rows K+1 / K+3
      const v2f b0 = *(const v2f*)&Bs[kp * B_ROW + (waveN + lm) * 2];
      const v2f b1 = *(const v2f*)&Bs[kp * B_ROW + (waveN + 16 + lm) * 2];
      acc00 = __builtin_amdgcn_wmma_f32_16x16x4_f32(false, a0, false, b0,
                                                    (short)0, acc00, false, false);
      acc01 = __builtin_amdgcn_wmma_f32_16x16x4_f32(false, a0, false, b1,
                                                    (short)0, acc01, false, false);
      acc10 = __builtin_amdgcn_wmma_f32_16x16x4_f32(false, a1, false, b0,
                                                    (short)0, acc10, false, false);
      acc11 = __builtin_amdgcn_wmma_f32_16x16x4_f32(false, a1, false, b1,
                                                    (short)0, acc11, false, false);
    }
    __syncthreads();
  }

  const float btot = beta + (betaPtr ? betaPtr[lb] : 0.0f);
  const int col0 = tileN + waveN + lm;
  const int col1 = col0 + 16;
#pragma unroll
  for (int r = 0; r < 8; ++r) {
    const int row0 = tileM + waveM + r + 8 * hw;
    const int row1 = row0 + 16;
    C[(long)row0 * MATN + col0] = alpha * acc00[r] + (row0 == col0 ? btot : 0.0f);
    C[(long)row0 * MATN + col1] = alpha * acc01[r] + (row0 == col1 ? btot : 0.0f);
    C[(long)row1 * MATN + col0] = alpha * acc10[r] + (row1 == col0 ? btot : 0.0f);
    C[(long)row1 * MATN + col1] = alpha * acc11[r] + (row1 == col1 ? btot : 0.0f);
  }
}

// ---------------------------------------------------------------------------
// Elementwise helpers (batched over blockIdx.y, 65536 elems / matrix)
// ---------------------------------------------------------------------------
__global__ __launch_bounds__(256) void k_identity(float* __restrict__ dst, float v) {
  const long b = blockIdx.y;
  const int  i = blockIdx.x * 256 + threadIdx.x;
  dst[b * MATE + i] = ((i >> 8) == (i & 255)) ? v : 0.0f;
}

// mode 0: dst = src * (1/s[b]);  mode 1: dst = src * rsqrt(s[b])
__global__ __launch_bounds__(256) void k_scale(const float* __restrict__ src,
                                               float* __restrict__ dst,
                                               const float* __restrict__ s, int mode) {
  const long b = blockIdx.y;
  const int  i = blockIdx.x * 256 + threadIdx.x;
  const float sv = s[b];
  const float f  = (mode == 0) ? (1.0f / sv) : rsqrtf(sv);
  dst[b * MATE + i] = src[b * MATE + i] * f;
}

__global__ __launch_bounds__(256) void k_subI(const float* __restrict__ src,
                                              float* __restrict__ dst) {
  const long b = blockIdx.y;
  const int  i = blockIdx.x * 256 + threadIdx.x;
  dst[b * MATE + i] = src[b * MATE + i] - (((i >> 8) == (i & 255)) ? 1.0f : 0.0f);
}

// Frobenius norm per matrix: s[b] = ||M_b||_F , lg[b] = log(s[b])
__global__ __launch_bounds__(256) void k_frob(const float* __restrict__ src,
                                              float* __restrict__ s,
                                              float* __restrict__ lg) {
  __shared__ float red[256];
  const float* M = src + (long)blockIdx.x * MATE;
  float sum = 0.0f;
  for (int r = 0; r < MATN; ++r) {
    const float v = M[(long)r * MATN + threadIdx.x];
    sum += v * v;
  }
  red[threadIdx.x] = sum;
  __syncthreads();
  for (int off = 128; off > 0; off >>= 1) {
    if (threadIdx.x < off) red[threadIdx.x] += red[threadIdx.x + off];
    __syncthreads();
  }
  if (threadIdx.x == 0) {
    const float fv = sqrtf(red[0]);
    s[blockIdx.x]  = fv;
    lg[blockIdx.x] = logf(fv);
  }
}

// ---------------------------------------------------------------------------
// Host orchestration (graph-capture safe: only launches on `stream`)
// ---------------------------------------------------------------------------
extern "C" void kernel_launch(void* const* d_in, const int* in_sizes, int n_in,
                              void* d_out, int out_size, void* d_ws, size_t ws_size,
                              hipStream_t stream) {
  (void)in_sizes; (void)n_in; (void)out_size;
  const float* inp = (const float*)d_in[0];   // (64, 256, 256) f32 SPD
  const float* wgt = (const float*)d_in[1];   // ( 8, 256, 256) f32 SPD
  float* out = (float*)d_out;                 // (64, 8, 256, 256) f32

  const int Cw = 8, P = 512;                  // pairs: p = b*8 + c
  const int NS_W = 14;                        // NS iters for W^{-1/2}
  const int NS_S = 10;                        // NS iters per sqrt level
  const int KR   = 5;                         // sqrt levels (inverse scaling)

  float* f = (float*)d_ws;
  // fixed weight-stage region: 48 matrices + scalars
  float* wY[2] = { f,             f + 8 * MATE };
  float* wZ[2] = { f + 16 * MATE, f + 24 * MATE };
  float* wT    = f + 32 * MATE;
  float* sri   = f + 40 * MATE;               // 8 matrices: W^{-1/2}
  float* wS    = f + 48 * MATE;
  float* wLg   = wS + 8;
  float* base  = f + 48 * MATE + 64;

  // chunk pairs so 5 working matrices/pair stay resident in the 192 MB L2
  long remain = (long)(ws_size / sizeof(float)) - (48 * MATE + 64);
  int chunk = (int)(remain / (5 * MATE + 2));
  if (chunk > 64) chunk = 64;                 // 5*64*256KB = 80 MB working set
  if (chunk > P)  chunk = P;
  if (chunk < 1)  chunk = 1;

  float* pS  = base;
  float* pLg = base + chunk;
  float* pb  = base + 2 * chunk;
  float* Yb[2] = { pb,                     pb + (long)chunk * MATE };
  float* Zb[2] = { pb + 2L * chunk * MATE, pb + 3L * chunk * MATE };
  float* T     = pb + 4L * chunk * MATE;

  // Horner coefficients for log(I+X) = X * sum_j (-1)^j X^j/(j+1), 8 terms
  const float cj[8] = { 1.0f, -0.5f, 1.0f / 3.0f, -0.25f,
                        0.2f, -1.0f / 6.0f, 1.0f / 7.0f, -0.125f };

  // ---- Stage 1: sri = W^{-1/2} (batch of 8) -------------------------------
  {
    dim3 gg(8, Cw), ge(256, Cw);
    k_frob    <<<Cw, 256, 0, stream>>>(wgt, wS, wLg);
    k_scale   <<<ge, 256, 0, stream>>>(wgt, wY[0], wS, 0);   // Y0 = W/s
    k_identity<<<ge, 256, 0, stream>>>(wZ[0], 1.0f);         // Z0 = I
    int y = 0;
    for (int it = 0; it < NS_W; ++it) {
      gemm256<<<gg, 256, 0, stream>>>(wZ[y], wY[y], wT, 0, 0, 0, -0.5f, 1.5f, nullptr);
      gemm256<<<gg, 256, 0, stream>>>(wY[y], wT, wY[1 - y], 0, 0, 0, 1.0f, 0.0f, nullptr);
      gemm256<<<gg, 256, 0, stream>>>(wT, wZ[y], wZ[1 - y], 0, 0, 0, 1.0f, 0.0f, nullptr);
      y ^= 1;
    }
    k_scale<<<ge, 256, 0, stream>>>(wZ[y], sri, wS, 1);      // sri = Z * s^{-1/2}
  }

  // ---- Stage 2: per-pair congruence + matrix log --------------------------
  for (int p0 = 0; p0 < P; p0 += chunk) {
    const int cn = (P - p0 < chunk) ? (P - p0) : chunk;
    dim3 gg(8, cn), ge(256, cn);

    // cong = sri[c] @ inp[b] @ sri[c]
    gemm256<<<gg, 256, 0, stream>>>(sri, inp, T, 1, 2, p0, 1.0f, 0.0f, nullptr);
    gemm256<<<gg, 256, 0, stream>>>(T, sri, Yb[0], 0, 1, p0, 1.0f, 0.0f, nullptr);

    // normalize: M^ = cong/s, remember log(s)
    k_frob <<<cn, 256, 0, stream>>>(Yb[0], pS, pLg);
    k_scale<<<ge, 256, 0, stream>>>(Yb[0], Yb[0], pS, 0);

    // KR repeated square roots via coupled Newton-Schulz
    int y = 0, z = 0;
    for (int lvl = 0; lvl < KR; ++lvl) {
      k_identity<<<ge, 256, 0, stream>>>(Zb[z], 1.0f);
      for (int it = 0; it < NS_S; ++it) {
        gemm256<<<gg, 256, 0, stream>>>(Zb[z], Yb[y], T, 0, 0, p0, -0.5f, 1.5f, nullptr);
        gemm256<<<gg, 256, 0, stream>>>(Yb[y], T, Yb[1 - y], 0, 0, p0, 1.0f, 0.0f, nullptr);
        gemm256<<<gg, 256, 0, stream>>>(T, Zb[z], Zb[1 - z], 0, 0, p0, 1.0f, 0.0f, nullptr);
        y ^= 1; z ^= 1;
      }
    }

    // X = M^{1/2^k} - I  (into T; Z buffers become Horner ping-pong)
    k_subI<<<ge, 256, 0, stream>>>(Yb[y], T);

    k_identity<<<ge, 256, 0, stream>>>(Zb[z], cj[7]);
    for (int j = 6; j >= 0; --j) {
      gemm256<<<gg, 256, 0, stream>>>(T, Zb[z], Zb[1 - z], 0, 0, p0, 1.0f, cj[j], nullptr);
      z ^= 1;
    }
    // out = 2^k * (X @ P) + log(s) * I   -> written straight to d_out
    gemm256<<<gg, 256, 0, stream>>>(T, Zb[z], out + (long)p0 * MATE, 0, 0, p0,
                                    (float)(1 << KR), 0.0f, pLg);
  }
}